// AladynSurvivalFixedKernelsAvgLoss_clust_logitInitgp_5514738008974
// MI455X (gfx1250) — compile-verified
//
#include <hip/hip_runtime.h>
#include <hip/hip_bf16.h>

// Problem constants (from reference)
#define PN 20000
#define PK 20
#define PT 50
#define PD 200
#define PEPS 1e-8f

typedef float v2f __attribute__((ext_vector_type(2)));
typedef float v8f __attribute__((ext_vector_type(8)));

// ---------------------------------------------------------------------------
// Kernel 1: theta = softmax(lambda_, axis=K).  One thread per (n,t) pair.
// t is the fastest axis, so a wave of consecutive i gives coalesced reads
// for each fixed k (stride-T gathers become 128B-contiguous across lanes).
// ---------------------------------------------------------------------------
__global__ __launch_bounds__(256) void theta_softmax_kernel(
    const float* __restrict__ lam, float* __restrict__ theta) {
  int i = blockIdx.x * 256 + threadIdx.x;
  if (i >= PN * PT) return;
  int n = i / PT;
  int t = i - n * PT;
  const float* p = lam + (size_t)n * (PK * PT) + t;
  float v[PK];
  float m = -3.402823466e38f;
#pragma unroll
  for (int k = 0; k < PK; ++k) {
    v[k] = p[(size_t)k * PT];
    m = fmaxf(m, v[k]);
  }
  float s = 0.f;
#pragma unroll
  for (int k = 0; k < PK; ++k) {
    v[k] = __expf(v[k] - m);
    s += v[k];
  }
  float inv = 1.f / s;
  float* q = theta + (size_t)n * (PK * PT) + t;
#pragma unroll
  for (int k = 0; k < PK; ++k) q[(size_t)k * PT] = v[k] * inv;
}

// ---------------------------------------------------------------------------
// Kernel 2: phi_prob = sigmoid(phi).  Trivial elementwise, 200K elements.
// ---------------------------------------------------------------------------
__global__ __launch_bounds__(256) void phi_sigmoid_kernel(
    const float* __restrict__ phi, float* __restrict__ out) {
  int i = blockIdx.x * 256 + threadIdx.x;
  if (i < PK * PD * PT) {
    float x = phi[i];
    out[i] = 1.f / (1.f + __expf(-x));
  }
}

// ---------------------------------------------------------------------------
// Kernel 3: pi[n,d,t] = sum_k theta[n,k,t] * phi_prob[k,d,t], clipped.
// Per fixed t this is a (16n x 20k) x (20k x 16d) GEMM tile done with five
// V_WMMA_F32_16X16X4_F32 steps.  One block per 16-row n-tile; 8 waves split
// the 50 t-slices; loop over 13 d-tiles (the 13th starts at d0 = D-16 = 184,
// overlapping the 12th with identical values -> no tail masking anywhere).
// Results staged through LDS so the dominant 800MB pi store is fully
// coalesced float4 traffic.
// ---------------------------------------------------------------------------
__global__ __launch_bounds__(256) void pi_wmma_kernel(
    const float* __restrict__ theta, const float* __restrict__ phip,
    float* __restrict__ pi) {
  __shared__ float sPi[16 * 16 * PT];  // 51.2 KB: [n_local][d_local][t]

  const int tid = threadIdx.x;
  const int wave = tid >> 5;
  const int lane = tid & 31;
  const int l16 = lane & 15;
  const int hi = lane >> 4;          // 0 for lanes 0-15, 1 for lanes 16-31
  const int khalf = hi << 1;         // ISA 7.12.2: V0 holds k, k+2 per half
  const int n0 = blockIdx.x * 16;

  // --- Preload A fragments (theta) into registers; reused for all d-tiles.
  // A 16x4 f32 layout: M = lane%16; V0 = K(4s+khalf), V1 = K(4s+khalf+1).
  v2f A[7][5];
#pragma unroll
  for (int i = 0; i < 7; ++i) {
    const int t = wave + 8 * i;      // wave-uniform
    if (t < PT) {
      const float* tb = theta + (size_t)(n0 + l16) * (PK * PT) + t;
#pragma unroll
      for (int s = 0; s < 5; ++s) {
        const int k = 4 * s + khalf;
        v2f a;
        a.x = tb[(size_t)k * PT];
        a.y = tb[(size_t)(k + 1) * PT];
        A[i][s] = a;
      }
    } else {
#pragma unroll
      for (int s = 0; s < 5; ++s) A[i][s] = (v2f){0.f, 0.f};
    }
  }

  for (int dt = 0; dt < 13; ++dt) {
    // Full 16-wide tile always; last tile overlaps the previous one by 8
    // columns (same block, identical recomputed values -> benign).
    const int d0 = (dt < 12) ? dt * 16 : (PD - 16);
    const int d = d0 + l16;

#pragma unroll
    for (int i = 0; i < 7; ++i) {
      const int t = wave + 8 * i;    // wave-uniform guard -> EXEC all-ones
      if (t < PT) {
        v8f c = {};
        const float* bb = phip + (size_t)d * PT + t;
#pragma unroll
        for (int s = 0; s < 5; ++s) {
          const int k = 4 * s + khalf;
          v2f B;
          B.x = bb[(size_t)k * (PD * PT)];
          B.y = bb[(size_t)(k + 1) * (PD * PT)];
          c = __builtin_amdgcn_wmma_f32_16x16x4_f32(
              /*neg_a=*/false, A[i][s], /*neg_b=*/false, B,
              /*c_mod=*/(short)0, c, /*reuse_a=*/false, /*reuse_b=*/false);
        }
        // C layout: VGPR r -> M = r (lanes 0-15) / r+8 (lanes 16-31); N = lane%16
#pragma unroll
        for (int r = 0; r < 8; ++r) {
          const int nl = r + (hi << 3);
          sPi[(nl * 16 + l16) * PT + t] = c[r];
        }
      }
    }
    __syncthreads();

    // --- Cooperative coalesced store: per n-row, [d0..d0+16) x [0..PT) is one
    // contiguous run in global memory AND in sPi -> straight float4 copy.
    const int rowq = 16 * PT / 4;              // 200 float4 per n-row
    const float4* s4 = (const float4*)sPi;
    for (int j = tid; j < 16 * rowq; j += 256) {
      const int n = j / rowq;
      const int r = j - n * rowq;
      float4 v = s4[j];
      v.x = fminf(fmaxf(v.x, PEPS), 1.f - PEPS);
      v.y = fminf(fmaxf(v.y, PEPS), 1.f - PEPS);
      v.z = fminf(fmaxf(v.z, PEPS), 1.f - PEPS);
      v.w = fminf(fmaxf(v.w, PEPS), 1.f - PEPS);
      float4* g = (float4*)(pi + (size_t)(n0 + n) * (PD * PT) + (size_t)d0 * PT);
      g[r] = v;
    }
    __syncthreads();
  }
}

// ---------------------------------------------------------------------------
extern "C" void kernel_launch(void* const* d_in, const int* in_sizes, int n_in,
                              void* d_out, int out_size, void* d_ws, size_t ws_size,
                              hipStream_t stream) {
  const float* lam = (const float*)d_in[0];   // [N,K,T]
  const float* phi = (const float*)d_in[1];   // [K,D,T]
  float* out = (float*)d_out;
  float* pi = out;                                   // N*D*T = 200,000,000
  float* theta = out + (size_t)PN * PD * PT;         // N*K*T =  20,000,000
  float* phip = theta + (size_t)PN * PK * PT;        // K*D*T =     200,000

  theta_softmax_kernel<<<(PN * PT + 255) / 256, 256, 0, stream>>>(lam, theta);
  phi_sigmoid_kernel<<<(PK * PD * PT + 255) / 256, 256, 0, stream>>>(phi, phip);
  pi_wmma_kernel<<<PN / 16, 256, 0, stream>>>(theta, phip, pi);
}